// Query_Weight_Network_84859963835017
// MI455X (gfx1250) — compile-verified
//
#include <hip/hip_runtime.h>
#include <hip/hip_bf16.h>

// Fused Query-Weight-Network kernel for gfx1250 (MI455X).
//
// out[r] = (||z3_r||^2 * <Xp1_r, Xp2_r>) /
//          max(||z3_r||^2 * ||Xp1_r|| * ||Xp2_r||, 1e-8)
// where z3 = relu(relu(relu(Xq@W1^T+b1)@W2^T+b2)@W3^T+b3).
//
// Memory-bound problem (37.7 MB X vs 0.9 GFLOP) -> single pass over X,
// f32 WMMA (V_WMMA_F32_16X16X4_F32) for the MLP, outer products collapsed
// algebraically (saves ~1.2 GB of intermediate traffic vs the reference).

#define EMB   768
#define L1N   128
#define L2N   64
#define L3N   32
#define BATCH 4096
#define KC    32      // K-chunk of W1 staged in LDS per iteration
#define WAVES 4       // waves per workgroup (wave32)
#define ROWS_PER_WG (WAVES * 16)

#define W1_STRIDE 36   // padded row stride (floats): conflict-free + 16B aligned
#define Z_STRIDE  132  // padded z-scratch row stride (floats): conflict-free

typedef __attribute__((ext_vector_type(2))) float v2f;
typedef __attribute__((ext_vector_type(4))) float v4f;
typedef __attribute__((ext_vector_type(8))) float v8f;

__global__ __launch_bounds__(128) void qwn_fused_kernel(
    const float* __restrict__ X,
    const float* __restrict__ W1, const float* __restrict__ b1,
    const float* __restrict__ W2, const float* __restrict__ b2,
    const float* __restrict__ W3, const float* __restrict__ b3,
    float* __restrict__ out)
{
    __shared__ float lds_w1[L1N * W1_STRIDE];       // 18 KB: W1 K-chunk
    __shared__ float zbuf[WAVES * 16 * Z_STRIDE];   // 33 KB: per-wave z scratch

    const int tid  = threadIdx.x;
    const int wave = tid >> 5;
    const int lane = tid & 31;
    const int n16  = lane & 15;   // N (or M) index within a 16-wide tile
    const int kh   = lane >> 4;   // K-half selector for A/B fragments
    const int tile = blockIdx.x * WAVES + wave;
    const int row0 = tile * 16;

    float* zw = &zbuf[wave * 16 * Z_STRIDE];

    const v8f vzero = {0.f, 0.f, 0.f, 0.f, 0.f, 0.f, 0.f, 0.f};

    // ---------------- GEMM1: z1 = relu(Xq @ W1^T + b1), 16x128 per wave ----
    v8f acc[8];
#pragma unroll
    for (int t = 0; t < 8; ++t) acc[t] = vzero;

    // A-frag source: lane holds row m=n16, K pair at k + 2*kh (16x4 A layout)
    const float* xrow = X + (size_t)(row0 + n16) * (3 * EMB);

    for (int kb = 0; kb < EMB; kb += KC) {
        // Stage W1[:, kb:kb+KC] into LDS: thread tid owns row n = tid.
        {
            const float* src = W1 + (size_t)tid * EMB + kb;
            float* dst = &lds_w1[tid * W1_STRIDE];
#pragma unroll
            for (int i = 0; i < KC / 4; ++i)
                *(v4f*)(dst + 4 * i) = *(const v4f*)(src + 4 * i);
        }
        if (kb + KC < EMB)
            __builtin_prefetch(W1 + (size_t)tid * EMB + kb + KC, 0, 0);
        __syncthreads();

#pragma unroll
        for (int k2 = 0; k2 < KC; k2 += 4) {
            v2f a = *(const v2f*)(xrow + kb + k2 + 2 * kh);
#pragma unroll
            for (int t = 0; t < 8; ++t) {
                const int n = n16 + 16 * t;
                v2f b = *(const v2f*)(&lds_w1[n * W1_STRIDE + k2 + 2 * kh]);
                acc[t] = __builtin_amdgcn_wmma_f32_16x16x4_f32(
                    false, a, false, b, (short)0, acc[t], false, false);
            }
        }
        __syncthreads();
    }

    // bias + relu; C layout: vgpr v -> row m = v + 8*kh, col n = n16 + 16t
#pragma unroll
    for (int t = 0; t < 8; ++t) {
        const int n = n16 + 16 * t;
        const float bias = b1[n];
#pragma unroll
        for (int v = 0; v < 8; ++v)
            zw[(v + 8 * kh) * Z_STRIDE + n] = fmaxf(acc[t][v] + bias, 0.0f);
    }
    __syncthreads();

    // ---------------- GEMM2: z2 = relu(z1 @ W2^T + b2), 16x64 ----
    v8f acc2[4];
#pragma unroll
    for (int t = 0; t < 4; ++t) acc2[t] = vzero;

    for (int k = 0; k < L1N; k += 4) {
        v2f a = *(const v2f*)(&zw[n16 * Z_STRIDE + k + 2 * kh]);
#pragma unroll
        for (int t = 0; t < 4; ++t) {
            const int n = n16 + 16 * t;
            v2f b = *(const v2f*)(W2 + (size_t)n * L1N + k + 2 * kh);
            acc2[t] = __builtin_amdgcn_wmma_f32_16x16x4_f32(
                false, a, false, b, (short)0, acc2[t], false, false);
        }
    }
    __syncthreads();
#pragma unroll
    for (int t = 0; t < 4; ++t) {
        const int n = n16 + 16 * t;
        const float bias = b2[n];
#pragma unroll
        for (int v = 0; v < 8; ++v)
            zw[(v + 8 * kh) * Z_STRIDE + n] = fmaxf(acc2[t][v] + bias, 0.0f);
    }
    __syncthreads();

    // ---------------- GEMM3: z3 = relu(z2 @ W3^T + b3), 16x32 ----
    v8f acc3[2];
#pragma unroll
    for (int t = 0; t < 2; ++t) acc3[t] = vzero;

    for (int k = 0; k < L2N; k += 4) {
        v2f a = *(const v2f*)(&zw[n16 * Z_STRIDE + k + 2 * kh]);
#pragma unroll
        for (int t = 0; t < 2; ++t) {
            const int n = n16 + 16 * t;
            v2f b = *(const v2f*)(W3 + (size_t)n * L2N + k + 2 * kh);
            acc3[t] = __builtin_amdgcn_wmma_f32_16x16x4_f32(
                false, a, false, b, (short)0, acc3[t], false, false);
        }
    }
    __syncthreads();
#pragma unroll
    for (int t = 0; t < 2; ++t) {
        const int n = n16 + 16 * t;
        const float bias = b3[n];
#pragma unroll
        for (int v = 0; v < 8; ++v)
            zw[(v + 8 * kh) * Z_STRIDE + n] = fmaxf(acc3[t][v] + bias, 0.0f);
    }
    __syncthreads();

    // ---------------- ||z3||^2 per row (row m = n16, both lane-halves) ----
    float n3sq = 0.0f;
#pragma unroll
    for (int n = 0; n < L3N; n += 4) {
        v4f z = *(const v4f*)(&zw[n16 * Z_STRIDE + n]);
        n3sq += z.x * z.x + z.y * z.y + z.z * z.z + z.w * z.w;
    }

    // ---------------- Xp1/Xp2 reductions: 2 lanes per row, 384 floats each --
    const float* p1 = X + (size_t)(row0 + n16) * (3 * EMB) + EMB + kh * (EMB / 2);
    const float* p2 = p1 + EMB;
    float s12 = 0.0f, s11 = 0.0f, s22 = 0.0f;
#pragma unroll 4
    for (int j = 0; j < EMB / 2; j += 4) {
        v4f u = *(const v4f*)(p1 + j);
        v4f w = *(const v4f*)(p2 + j);
        s12 += u.x * w.x + u.y * w.y + u.z * w.z + u.w * w.w;
        s11 += u.x * u.x + u.y * u.y + u.z * u.z + u.w * u.w;
        s22 += w.x * w.x + w.y * w.y + w.z * w.z + w.w * w.w;
    }
    s12 += __shfl_xor(s12, 16, 32);
    s11 += __shfl_xor(s11, 16, 32);
    s22 += __shfl_xor(s22, 16, 32);

    if (lane < 16) {
        const float num = n3sq * s12;
        const float den = fmaxf(n3sq * sqrtf(s11) * sqrtf(s22), 1e-8f);
        out[row0 + lane] = num / den;
    }
}

extern "C" void kernel_launch(void* const* d_in, const int* in_sizes, int n_in,
                              void* d_out, int out_size, void* d_ws, size_t ws_size,
                              hipStream_t stream) {
    (void)in_sizes; (void)n_in; (void)out_size; (void)d_ws; (void)ws_size;
    const float* X  = (const float*)d_in[0];
    const float* W1 = (const float*)d_in[1];
    const float* b1 = (const float*)d_in[2];
    const float* W2 = (const float*)d_in[3];
    const float* b2 = (const float*)d_in[4];
    const float* W3 = (const float*)d_in[5];
    const float* b3 = (const float*)d_in[6];
    float* out = (float*)d_out;

    dim3 grid(BATCH / ROWS_PER_WG);  // 64 workgroups
    dim3 block(WAVES * 32);          // 128 threads = 4 wave32
    qwn_fused_kernel<<<grid, block, 0, stream>>>(X, W1, b1, W2, b2, W3, b3, out);
}